// GeometryPreservingDiffusionLoss_56332791054448
// MI455X (gfx1250) — compile-verified
//
#include <hip/hip_runtime.h>
#include <math.h>

// Problem constants (match reference setup_inputs)
#define B_   8
#define N_   4096
#define C_   256
#define D_   512
#define KL   16
#define KS   4
#define NBIN 36

// Accumulator slots (floats at start of d_ws)
#define A_DIFF   0
#define A_MSE    1
#define A_OV     2
#define A_NV     3
#define A_ACT    4
#define A_STYLE  5
#define A_LOCAL  6
#define A_CENTER 7
#define A_RANGE  8
#define A_COV    9
#define A_DISTM  10
#define A_DISTS  11
#define A_LIDAR  12
#define A_SS     16   // + b : per-batch sum of nd_mean
#define A_SQ     24   // + b : per-batch sumsq of nd_mean
#define ACC_N    128

typedef __attribute__((ext_vector_type(2))) float v2f;
typedef __attribute__((ext_vector_type(8))) float v8f;
typedef __attribute__((ext_vector_type(4))) int   v4i;

#define AS1 __attribute__((address_space(1)))
#define AS3 __attribute__((address_space(3)))

#if defined(__has_builtin)
#if __has_builtin(__builtin_amdgcn_global_load_async_to_lds_b128)
#define HAVE_ASYNC_LDS 1
#endif
#endif

// ---------------- block reduction helpers (blockDim.x == 256, 8 waves) ----
__device__ __forceinline__ float warpSum(float v) {
#pragma unroll
  for (int o = 16; o > 0; o >>= 1) v += __shfl_xor(v, o, 32);
  return v;
}
__device__ __forceinline__ float warpMin(float v) {
#pragma unroll
  for (int o = 16; o > 0; o >>= 1) v = fminf(v, __shfl_xor(v, o, 32));
  return v;
}
__device__ __forceinline__ float warpMax(float v) {
#pragma unroll
  for (int o = 16; o > 0; o >>= 1) v = fmaxf(v, __shfl_xor(v, o, 32));
  return v;
}
__device__ __forceinline__ float blockSum(float v, float* red) {
  int lane = threadIdx.x & 31, w = threadIdx.x >> 5;
  v = warpSum(v);
  __syncthreads();
  if (lane == 0) red[w] = v;
  __syncthreads();
  float r = (lane < 8) ? red[lane] : 0.0f;
#pragma unroll
  for (int o = 4; o > 0; o >>= 1) r += __shfl_xor(r, o, 32);
  return __shfl(r, 0, 32);
}
__device__ __forceinline__ float blockMin(float v, float* red) {
  int lane = threadIdx.x & 31, w = threadIdx.x >> 5;
  v = warpMin(v);
  __syncthreads();
  if (lane == 0) red[w] = v;
  __syncthreads();
  float r = (lane < 8) ? red[lane] : 3.0e38f;
#pragma unroll
  for (int o = 4; o > 0; o >>= 1) r = fminf(r, __shfl_xor(r, o, 32));
  return __shfl(r, 0, 32);
}
__device__ __forceinline__ float blockMax(float v, float* red) {
  int lane = threadIdx.x & 31, w = threadIdx.x >> 5;
  v = warpMax(v);
  __syncthreads();
  if (lane == 0) red[w] = v;
  __syncthreads();
  float r = (lane < 8) ? red[lane] : -3.0e38f;
#pragma unroll
  for (int o = 4; o > 0; o >>= 1) r = fmaxf(r, __shfl_xor(r, o, 32));
  return __shfl(r, 0, 32);
}

// ---------------- prep: zero accumulators, build padded points + sq norms --
__global__ __launch_bounds__(256) void prep_kernel(
    const float* __restrict__ orig, const float* __restrict__ gen,
    float* __restrict__ sqo, float* __restrict__ sqg,
    float4* __restrict__ p4o, float4* __restrict__ p4g,
    float* __restrict__ acc) {
  int i = blockIdx.x * 256 + threadIdx.x;
  if (blockIdx.x == 0 && threadIdx.x < ACC_N) acc[threadIdx.x] = 0.0f;
  if (i < B_ * N_) {
    float ox = orig[3 * i], oy = orig[3 * i + 1], oz = orig[3 * i + 2];
    float gx = gen[3 * i],  gy = gen[3 * i + 1],  gz = gen[3 * i + 2];
    p4o[i] = make_float4(ox, oy, oz, 0.0f);
    p4g[i] = make_float4(gx, gy, gz, 0.0f);
    sqo[i] = ox * ox + oy * oy + oz * oz;
    sqg[i] = gx * gx + gy * gy + gz * gz;
  }
}

// ---------------- diffusion MSE ------------------------------------------
__global__ __launch_bounds__(256) void diff_kernel(
    const float* __restrict__ a, const float* __restrict__ b,
    float* __restrict__ acc, int n) {
  __shared__ float red[8];
  float s = 0.0f;
  for (int i = blockIdx.x * 256 + threadIdx.x; i < n; i += gridDim.x * 256) {
    float d = a[i] - b[i];
    s += d * d;
  }
  s = blockSum(s, red);
  if (threadIdx.x == 0) atomicAdd(&acc[A_DIFF], s);
}

// ---------------- content loss: one block per (b,c) row of length N -------
__global__ __launch_bounds__(256) void content_kernel(
    const float* __restrict__ co, const float* __restrict__ cn,
    float* __restrict__ acc) {
  __shared__ float red[8];
  size_t row = blockIdx.x;
  const float* o = co + row * (size_t)N_;
  const float* n = cn + row * (size_t)N_;
  float smse = 0, so = 0, soo = 0, sn = 0, snn = 0, sab = 0;
  for (int i = threadIdx.x; i < N_; i += 256) {
    float ov = o[i], nv = n[i];
    float d = nv - ov;
    smse += d * d;
    so += ov; soo += ov * ov;
    sn += nv; snn += nv * nv;
    sab += fabsf(ov);
  }
  smse = blockSum(smse, red);
  so = blockSum(so, red);   soo = blockSum(soo, red);
  sn = blockSum(sn, red);   snn = blockSum(snn, red);
  sab = blockSum(sab, red);
  if (threadIdx.x == 0) {
    const float N = (float)N_;
    float varo = (soo - so * so / N) / (N - 1.0f);
    float varn = (snn - sn * sn / N) / (N - 1.0f);
    atomicAdd(&acc[A_MSE], smse);
    atomicAdd(&acc[A_OV], varo);
    atomicAdd(&acc[A_NV], varn);
    atomicAdd(&acc[A_ACT], sab);
  }
}

// ---------------- style loss: one block per batch row --------------------
__global__ __launch_bounds__(256) void style_kernel(
    const float* __restrict__ src, const float* __restrict__ tgt,
    float* __restrict__ acc) {
  __shared__ float red[8];
  const float* a = src + blockIdx.x * D_;
  const float* b = tgt + blockIdx.x * D_;
  float sab = 0, saa = 0, sbb = 0;
  for (int i = threadIdx.x; i < D_; i += 256) {
    sab += a[i] * b[i]; saa += a[i] * a[i]; sbb += b[i] * b[i];
  }
  sab = blockSum(sab, red);
  saa = blockSum(saa, red);
  sbb = blockSum(sbb, red);
  if (threadIdx.x == 0) {
    float na = sqrtf(fmaxf(saa, 1e-12f));
    float nb = sqrtf(fmaxf(sbb, 1e-12f));
    float cosv = sab / fmaxf(na * nb, 1e-8f);
    atomicAdd(&acc[A_STYLE], fabsf(cosv - 0.5f));
  }
}

// ---------------- shape loss: one block per batch ------------------------
__global__ __launch_bounds__(256) void shape_kernel(
    const float* __restrict__ gen, const float* __restrict__ orig,
    float* __restrict__ acc) {
  __shared__ float red[8];
  const float* G = gen + blockIdx.x * N_ * 3;
  const float* O = orig + blockIdx.x * N_ * 3;
  const float N = (float)N_;
  float sg[3] = {0, 0, 0}, so[3] = {0, 0, 0};
  float mng[3] = {3e38f, 3e38f, 3e38f}, mxg[3] = {-3e38f, -3e38f, -3e38f};
  float mno[3] = {3e38f, 3e38f, 3e38f}, mxo[3] = {-3e38f, -3e38f, -3e38f};
  for (int i = threadIdx.x; i < N_; i += 256) {
#pragma unroll
    for (int a = 0; a < 3; ++a) {
      float gv = G[3 * i + a], ov = O[3 * i + a];
      sg[a] += gv; so[a] += ov;
      mng[a] = fminf(mng[a], gv); mxg[a] = fmaxf(mxg[a], gv);
      mno[a] = fminf(mno[a], ov); mxo[a] = fmaxf(mxo[a], ov);
    }
  }
  float gm[3], om[3], cenL = 0, rngL = 0;
#pragma unroll
  for (int a = 0; a < 3; ++a) {
    sg[a] = blockSum(sg[a], red);  so[a] = blockSum(so[a], red);
    mng[a] = blockMin(mng[a], red); mxg[a] = blockMax(mxg[a], red);
    mno[a] = blockMin(mno[a], red); mxo[a] = blockMax(mxo[a], red);
    gm[a] = sg[a] / N; om[a] = so[a] / N;
    float dc = gm[a] - om[a];
    cenL += dc * dc;
    float dr = (mxg[a] - mng[a]) - (mxo[a] - mno[a]);
    rngL += dr * dr;
  }
  // pass 2: covariance + centered-distance stats
  float gc[6] = {0, 0, 0, 0, 0, 0}, oc[6] = {0, 0, 0, 0, 0, 0};
  float sgd = 0, sgd2 = 0, sod = 0, sod2 = 0;
  for (int i = threadIdx.x; i < N_; i += 256) {
    float gx = G[3 * i] - gm[0], gy = G[3 * i + 1] - gm[1], gz = G[3 * i + 2] - gm[2];
    float ox = O[3 * i] - om[0], oy = O[3 * i + 1] - om[1], oz = O[3 * i + 2] - om[2];
    gc[0] += gx * gx; gc[1] += gx * gy; gc[2] += gx * gz;
    gc[3] += gy * gy; gc[4] += gy * gz; gc[5] += gz * gz;
    oc[0] += ox * ox; oc[1] += ox * oy; oc[2] += ox * oz;
    oc[3] += oy * oy; oc[4] += oy * oz; oc[5] += oz * oz;
    float gd = sqrtf(fmaxf(gx * gx + gy * gy + gz * gz, 1e-12f));
    float od = sqrtf(fmaxf(ox * ox + oy * oy + oz * oz, 1e-12f));
    sgd += gd; sgd2 += gd * gd; sod += od; sod2 += od * od;
  }
#pragma unroll
  for (int k = 0; k < 6; ++k) {
    gc[k] = blockSum(gc[k], red);
    oc[k] = blockSum(oc[k], red);
  }
  sgd = blockSum(sgd, red); sgd2 = blockSum(sgd2, red);
  sod = blockSum(sod, red); sod2 = blockSum(sod2, red);
  if (threadIdx.x == 0) {
    float inv = 1.0f / (N - 1.0f);
    float fro = 0.0f;
#pragma unroll
    for (int k = 0; k < 6; ++k) {
      float d = (gc[k] - oc[k]) * inv;
      float w = (k == 1 || k == 2 || k == 4) ? 2.0f : 1.0f;  // off-diag twice
      fro += w * d * d;
    }
    fro = sqrtf(fmaxf(fro, 1e-12f));
    float gdm = sgd / N, odm = sod / N;
    float gds = sqrtf(fmaxf((sgd2 - sgd * sgd / N) * inv, 0.0f));
    float ods = sqrtf(fmaxf((sod2 - sod * sod / N) * inv, 0.0f));
    atomicAdd(&acc[A_CENTER], cenL);
    atomicAdd(&acc[A_RANGE], rngL);
    atomicAdd(&acc[A_COV], fro);
    atomicAdd(&acc[A_DISTM], (gdm - odm) * (gdm - odm));
    atomicAdd(&acc[A_DISTS], (gds - ods) * (gds - ods));
  }
}

// ---------------- KNN kernel: WMMA 16x16x4 f32 pairwise dot tiles --------
// Each wave owns 16 rows; loops over 256 column tiles. Two WMMAs per tile
// (orig dot -> top-16 with indices, gen dot -> top-4 values). Per-batch
// squared norms are staged in LDS via GLOBAL_LOAD_ASYNC_TO_LDS_B128 (falls
// back to plain cooperative copies). Tile-loop synchronization is wave-local
// (DS ops of a wave are in-order; s_wait_dscnt 0 + memory clobber).
__global__ __launch_bounds__(256) void knn_kernel(
    const float4* __restrict__ p4o, const float4* __restrict__ p4g,
    const float* __restrict__ sqo, const float* __restrict__ sqg,
    float* __restrict__ acc) {
  __shared__ float sqOs[N_];
  __shared__ float sqGs[N_];
  __shared__ float ldsO[8][16][17];
  __shared__ float ldsG[8][16][17];
  __shared__ float stage[8][16][4];
  const int b = blockIdx.y;
  const int w = threadIdx.x >> 5, lane = threadIdx.x & 31;
  const int M = lane & 15;
  const int row = blockIdx.x * 128 + w * 16 + M;   // batch-local row index
  const int base = b * N_;

  // Stage per-batch squared-norm arrays into LDS.
  {
    const float* go = sqo + base;
    const float* gg = sqg + base;
#ifdef HAVE_ASYNC_LDS
    for (int k = threadIdx.x * 4; k < N_; k += 256 * 4) {
      __builtin_amdgcn_global_load_async_to_lds_b128(
          (AS1 v4i*)(go + k), (AS3 v4i*)&sqOs[k], 0, 0);
      __builtin_amdgcn_global_load_async_to_lds_b128(
          (AS1 v4i*)(gg + k), (AS3 v4i*)&sqGs[k], 0, 0);
    }
    asm volatile("s_wait_asynccnt 0x0" ::: "memory");
#else
    for (int k = threadIdx.x; k < N_; k += 256) {
      sqOs[k] = go[k];
      sqGs[k] = gg[k];
    }
#endif
  }
  __syncthreads();

  const float4 po = p4o[base + row];
  const float4 pg = p4g[base + row];
  // A fragment (16x4 f32): VGPR0 = K0 (lanes 0-15) / K2 (lanes 16-31),
  //                        VGPR1 = K1 (lanes 0-15) / K3 (lanes 16-31)
  v2f aO, aG;
  if (lane < 16) { aO[0] = po.x; aO[1] = po.y; aG[0] = pg.x; aG[1] = pg.y; }
  else           { aO[0] = po.z; aO[1] = 0.f;  aG[0] = pg.z; aG[1] = 0.f;  }

  float dO[KL]; int iO[KL]; float dG[KS];
#pragma unroll
  for (int k = 0; k < KL; ++k) { dO[k] = 3.0e38f; iO[k] = 0; }
#pragma unroll
  for (int k = 0; k < KS; ++k) dG[k] = 3.0e38f;

  const float srO = sqOs[row];
  const float srG = sqGs[row];
  const int colHalf = (lane >> 4) * 8;   // lanes 0-15: cols 0-7, 16-31: 8-15

  for (int jt = 0; jt < N_ / 16; ++jt) {
    const int colBase = jt * 16;
    if (jt + 1 < N_ / 16) {
      __builtin_prefetch(&p4o[base + colBase + 16 + M], 0, 1);
      __builtin_prefetch(&p4g[base + colBase + 16 + M], 0, 1);
    }
    const float4 qo = p4o[base + colBase + M];
    const float4 qg = p4g[base + colBase + M];
    // B fragment (4x16 f32): VGPR0 = row K0 (lanes 0-15) / K1 (lanes 16-31),
    //                        VGPR1 = row K2 (lanes 0-15) / K3 (lanes 16-31)
    v2f bO, bG;
    if (lane < 16) { bO[0] = qo.x; bO[1] = qo.z; bG[0] = qg.x; bG[1] = qg.z; }
    else           { bO[0] = qo.y; bO[1] = 0.f;  bG[0] = qg.y; bG[1] = 0.f;  }
    v8f cO = {}; v8f cG = {};
    cO = __builtin_amdgcn_wmma_f32_16x16x4_f32(false, aO, false, bO,
                                               (short)0, cO, false, false);
    cG = __builtin_amdgcn_wmma_f32_16x16x4_f32(false, aG, false, bG,
                                               (short)0, cG, false, false);
    const int mOff = (lane < 16) ? 0 : 8;
#pragma unroll
    for (int p = 0; p < 8; ++p) {
      ldsO[w][mOff + p][M] = cO[p];
      ldsG[w][mOff + p][M] = cG[p];
    }
    // Wave-local ordering: all DS stores complete before we read the tile.
    asm volatile("s_wait_dscnt 0x0" ::: "memory");
#pragma unroll
    for (int cc = 0; cc < 8; ++cc) {
      const int c = colHalf + cc;
      const int j = colBase + c;
      float d2o = srO + sqOs[j] - 2.0f * ldsO[w][M][c];
      float d2g = srG + sqGs[j] - 2.0f * ldsG[w][M][c];
      d2o = fmaxf(d2o, 0.0f);
      d2g = fmaxf(d2g, 0.0f);
      if (j == row) { d2o += 1.0e10f; d2g += 1.0e10f; }
      if (d2o < dO[KL - 1]) {           // fully-unrolled sorted insert
        float dv = d2o; int iv = j;
#pragma unroll
        for (int k = 0; k < KL; ++k) {
          bool sw = dv < dO[k];
          float td = dO[k]; int ti = iO[k];
          if (sw) { dO[k] = dv; iO[k] = iv; dv = td; iv = ti; }
        }
      }
      if (d2g < dG[KS - 1]) {
        float dv = d2g;
#pragma unroll
        for (int k = 0; k < KS; ++k) {
          bool sw = dv < dG[k];
          float td = dG[k];
          if (sw) { dG[k] = dv; dv = td; }
        }
      }
    }
    // Keep next iteration's tile stores below this iteration's reads.
    asm volatile("" ::: "memory");
  }

  // merge the two column-half lists per row via LDS (block-wide barrier ok:
  // all waves have finished their private tile loops)
  __syncthreads();
  if (lane >= 16) {
#pragma unroll
    for (int k = 0; k < KL; ++k) {
      ldsO[w][M][k] = dO[k];
      ldsG[w][M][k] = __int_as_float(iO[k]);
    }
#pragma unroll
    for (int k = 0; k < KS; ++k) stage[w][M][k] = dG[k];
  }
  __syncthreads();
  if (lane < 16) {
#pragma unroll
    for (int k = 0; k < KL; ++k) {
      float dv = ldsO[w][M][k]; int iv = __float_as_int(ldsG[w][M][k]);
      if (dv < dO[KL - 1]) {
#pragma unroll
        for (int q = 0; q < KL; ++q) {
          bool sw = dv < dO[q];
          float td = dO[q]; int ti = iO[q];
          if (sw) { dO[q] = dv; iO[q] = iv; dv = td; iv = ti; }
        }
      }
    }
#pragma unroll
    for (int k = 0; k < KS; ++k) {
      float dv = stage[w][M][k];
      if (dv < dG[KS - 1]) {
#pragma unroll
        for (int q = 0; q < KS; ++q) {
          bool sw = dv < dG[q];
          float td = dG[q];
          if (sw) { dG[q] = dv; dv = td; }
        }
      }
    }
    // local loss: sum over 16 nearest (orig) neighbors of
    // ||(gen[j]-gen[i]) - (orig[j]-orig[i])||^2   (j is batch-local index)
    const float4 gi = p4g[base + row];
    const float4 oi = p4o[base + row];
    float lsum = 0.0f;
#pragma unroll
    for (int k = 0; k < KL; ++k) {
      const int j = iO[k];
      const float4 gj = p4g[base + j];
      const float4 oj = p4o[base + j];
      float dx = (gj.x - gi.x) - (oj.x - oi.x);
      float dy = (gj.y - gi.y) - (oj.y - oi.y);
      float dz = (gj.z - gi.z) - (oj.z - oi.z);
      lsum += dx * dx + dy * dy + dz * dz;
    }
    float nd = 0.0f;
#pragma unroll
    for (int k = 0; k < KS; ++k) nd += sqrtf(fmaxf(dG[k], 1e-12f));
    nd *= 0.25f;
    atomicAdd(&acc[A_LOCAL], lsum);
    atomicAdd(&acc[A_SS + b], nd);
    atomicAdd(&acc[A_SQ + b], nd * nd);
  }
}

// ---------------- lidar loss: one block per batch ------------------------
__global__ __launch_bounds__(256) void lidar_kernel(
    const float* __restrict__ orig, const float* __restrict__ gen,
    float* __restrict__ acc) {
  __shared__ float thO[N_];
  __shared__ float thG[N_];
  __shared__ float hO[NBIN];
  __shared__ float hG[NBIN];
  __shared__ float selO[101][3];
  __shared__ float selG[101][3];
  __shared__ float red[8];
  const int t = threadIdx.x;
  const float* O = orig + blockIdx.x * N_ * 3;
  const float* G = gen + blockIdx.x * N_ * 3;
  const float PIF = 3.14159265358979f;
  const float scale = (float)NBIN / (2.0f * PIF);
  if (t < NBIN) { hO[t] = 0.0f; hG[t] = 0.0f; }
  __syncthreads();
  float szO = 0, szzO = 0, srO = 0, srrO = 0;
  float szG = 0, szzG = 0, srG = 0, srrG = 0;
  for (int i = t; i < N_; i += 256) {
    float ox = O[3 * i], oy = O[3 * i + 1], oz = O[3 * i + 2];
    float gx = G[3 * i], gy = G[3 * i + 1], gz = G[3 * i + 2];
    float ro = fmaxf(sqrtf(fmaxf(ox * ox + oy * oy, 1e-12f)), 1e-6f);
    float rg = fmaxf(sqrtf(fmaxf(gx * gx + gy * gy, 1e-12f)), 1e-6f);
    float to = atan2f(oy, ox);
    float tg = atan2f(gy, gx);
    thO[i] = to; thG[i] = tg;
    szO += oz; szzO += oz * oz; srO += ro; srrO += ro * ro;
    szG += gz; szzG += gz * gz; srG += rg; srrG += rg * rg;
    int io = min(max((int)floorf((to + PIF) * scale), 0), NBIN - 1);
    int ig = min(max((int)floorf((tg + PIF) * scale), 0), NBIN - 1);
    atomicAdd(&hO[io], 1.0f);
    atomicAdd(&hG[ig], 1.0f);
  }
  __syncthreads();
  szO = blockSum(szO, red); szzO = blockSum(szzO, red);
  srO = blockSum(srO, red); srrO = blockSum(srrO, red);
  szG = blockSum(szG, red); szzG = blockSum(szzG, red);
  srG = blockSum(srG, red); srrG = blockSum(srrG, red);
  // stable rank selection of the 100 smallest thetas
  for (int i = t; i < N_; i += 256) {
    float to = thO[i], tg = thG[i];
    int ro = 0, rg = 0;
    for (int j = 0; j < N_; ++j) {
      float vo = thO[j], vg = thG[j];
      ro += (vo < to) || (vo == to && j < i);
      rg += (vg < tg) || (vg == tg && j < i);
    }
    if (ro < 100) {
      selO[ro][0] = O[3 * i]; selO[ro][1] = O[3 * i + 1]; selO[ro][2] = O[3 * i + 2];
    }
    if (rg < 100) {
      selG[rg][0] = G[3 * i]; selG[rg][1] = G[3 * i + 1]; selG[rg][2] = G[3 * i + 2];
    }
  }
  __syncthreads();
  float csum = 0.0f;
  if (t < 99) {
    float dx = selO[t + 1][0] - selO[t][0];
    float dy = selO[t + 1][1] - selO[t][1];
    float dz = selO[t + 1][2] - selO[t][2];
    float don = fmaxf(sqrtf(fmaxf(dx * dx + dy * dy + dz * dz, 1e-12f)), 1e-6f);
    dx = selG[t + 1][0] - selG[t][0];
    dy = selG[t + 1][1] - selG[t][1];
    dz = selG[t + 1][2] - selG[t][2];
    float dgn = fmaxf(sqrtf(fmaxf(dx * dx + dy * dy + dz * dz, 1e-12f)), 1e-6f);
    csum = fabsf((dgn - don) / (don + 1e-6f));
  }
  csum = blockSum(csum, red) / 99.0f;
  float asum = 0.0f;
  if (t < NBIN) {
    float ho = hO[t] / ((float)N_ + 1e-8f);
    float hg = hG[t] / ((float)N_ + 1e-8f);
    asum = (hg - ho) * (hg - ho);
  }
  asum = blockSum(asum, red) / (float)NBIN;
  if (t == 0) {
    const float N = (float)N_;
    float inv = 1.0f / (N - 1.0f);
    float ozm = szO / N, gzm = szG / N;
    float ozs = sqrtf(fmaxf((szzO - szO * szO / N) * inv, 0.0f));
    float gzs = sqrtf(fmaxf((szzG - szG * szG / N) * inv, 0.0f));
    float orm = srO / N, grm = srG / N;
    float ors = sqrtf(fmaxf((srrO - srO * srO / N) * inv, 0.0f));
    float grs = sqrtf(fmaxf((srrG - srG * srG / N) * inv, 0.0f));
    float vert = fabsf(gzm - ozm) + fabsf(gzs - ozs);
    float radial = (fabsf(grm - orm) + fabsf(grs - ors)) * 0.1f;
    float lidar_b = asum + vert * 0.1f + csum * 0.5f + radial * 0.5f;
    atomicAdd(&acc[A_LIDAR], lidar_b);
  }
}

// ---------------- finalize ------------------------------------------------
__global__ void finalize_kernel(const float* __restrict__ acc,
                                float* __restrict__ out) {
  if (threadIdx.x != 0 || blockIdx.x != 0) return;
  const float B = (float)B_, N = (float)N_, C = (float)C_;
  float diff = acc[A_DIFF] / (B * N * 3.0f);
  float mse = acc[A_MSE] / (B * C * N);
  float ov = acc[A_OV] / (B * C);
  float nv = acc[A_NV] / (B * C);
  float act = acc[A_ACT] / (B * C * N);
  float content = mse + fmaxf(0.1f - ov, 0.0f) + fmaxf(0.1f - nv, 0.0f) +
                  fmaxf(1.0f - act, 0.0f) * 0.1f;
  float style = acc[A_STYLE] / B;
  float shape = acc[A_CENTER] / (B * 3.0f) + 0.5f * acc[A_RANGE] / (B * 3.0f) +
                0.1f * acc[A_COV] / B +
                0.5f * (acc[A_DISTM] / B + acc[A_DISTS] / B);
  float local = acc[A_LOCAL] / (B * N * (float)KL * 3.0f);
  float smooth = 0.0f;
  for (int b = 0; b < B_; ++b) {
    float s = acc[A_SS + b], q = acc[A_SQ + b];
    float var = (q - s * s / N) / (N - 1.0f);
    smooth += sqrtf(fmaxf(var, 0.0f));
  }
  smooth /= B;
  float lidar = acc[A_LIDAR] / B;
  out[0] = 1.0f * diff + 2.0f * content + 2.0f * shape + 1.0f * local +
           0.5f * smooth + 0.5f * lidar + 0.02f * style;
}

// ---------------- launch --------------------------------------------------
extern "C" void kernel_launch(void* const* d_in, const int* in_sizes, int n_in,
                              void* d_out, int out_size, void* d_ws,
                              size_t ws_size, hipStream_t stream) {
  const float* pred = (const float*)d_in[0];
  const float* targ = (const float*)d_in[1];
  const float* gen  = (const float*)d_in[2];
  const float* orig = (const float*)d_in[3];
  const float* co   = (const float*)d_in[4];
  const float* cn   = (const float*)d_in[5];
  const float* ss   = (const float*)d_in[6];
  const float* st   = (const float*)d_in[7];
  float* ws = (float*)d_ws;
  // workspace layout (floats): acc[128] | sqo[B*N] | sqg[B*N] | p4o | p4g
  float* acc = ws;
  float* sqo = ws + ACC_N;
  float* sqg = sqo + B_ * N_;
  float4* p4o = (float4*)(sqg + B_ * N_);
  float4* p4g = p4o + B_ * N_;

  prep_kernel<<<(B_ * N_ + 255) / 256, 256, 0, stream>>>(orig, gen, sqo, sqg,
                                                         p4o, p4g, acc);
  diff_kernel<<<96, 256, 0, stream>>>(pred, targ, acc, B_ * N_ * 3);
  content_kernel<<<B_ * C_, 256, 0, stream>>>(co, cn, acc);
  style_kernel<<<B_, 256, 0, stream>>>(ss, st, acc);
  shape_kernel<<<B_, 256, 0, stream>>>(gen, orig, acc);
  knn_kernel<<<dim3(N_ / 128, B_), 256, 0, stream>>>(p4o, p4g, sqo, sqg, acc);
  lidar_kernel<<<B_, 256, 0, stream>>>(orig, gen, acc);
  finalize_kernel<<<1, 32, 0, stream>>>(acc, (float*)d_out);
}